// RGCN_90374701843210
// MI455X (gfx1250) — compile-verified
//
#include <hip/hip_runtime.h>

typedef __attribute__((ext_vector_type(2))) float v2f;
typedef __attribute__((ext_vector_type(8))) float v8f;

#define N_NODES 30000
#define R_REL   16
#define B_BASES 30
#define E_EDGES 960000
#define D_IN    128
#define D_HID   128
#define D_OUT   64
#define D_ADR   400

// ---------------- graph aggregation ----------------

__global__ void rgcn_count_kernel(const int* __restrict__ dst,
                                  const int* __restrict__ et,
                                  float* __restrict__ cnt, int E) {
  int e = blockIdx.x * blockDim.x + threadIdx.x;
  if (e < E) {
    int seg = dst[e] * R_REL + et[e];
    atomicAdd(&cnt[seg], 1.0f);
  }
}

// 256 threads = 2 edges x 128 features
__global__ void rgcn_scatter_kernel(const float* __restrict__ feat,
                                    const int* __restrict__ src,
                                    const int* __restrict__ dst,
                                    const int* __restrict__ et,
                                    float* __restrict__ agg, int E) {
  int e = blockIdx.x * 2 + (threadIdx.x >> 7);
  int f = threadIdx.x & 127;
  if (e < E) {
    int s   = src[e];
    int seg = dst[e] * R_REL + et[e];
    atomicAdd(&agg[(size_t)seg * 128 + f], feat[(size_t)s * 128 + f]);
  }
}

__global__ void rgcn_mean_kernel(float* __restrict__ agg,
                                 const float* __restrict__ cnt, size_t total) {
  size_t i = (size_t)blockIdx.x * blockDim.x + threadIdx.x;
  if (i < total) {
    float c = cnt[i >> 7];
    if (c > 1.0f) agg[i] = agg[i] / c;
  }
}

// ---------------- weight stacking: W[(R+1)*DI, DO] = [comp x basis ; root] ----------------

__global__ void rgcn_wstack_kernel(const float* __restrict__ comp,   // [R,B]
                                   const float* __restrict__ basis,  // [B,DI,DO]
                                   const float* __restrict__ root,   // [DI,DO]
                                   float* __restrict__ W, int DI, int DO) {
  int t = blockIdx.x * blockDim.x + threadIdx.x;
  int tot = (R_REL + 1) * DI * DO;
  if (t >= tot) return;
  int o = t % DO;
  int rest = t / DO;
  int i = rest % DI;
  int r = rest / DI;
  float v;
  if (r < R_REL) {
    v = 0.0f;
    for (int b = 0; b < B_BASES; ++b)
      v += comp[r * B_BASES + b] * basis[((size_t)b * DI + i) * DO + o];
  } else {
    v = root[(size_t)i * DO + o];
  }
  W[(size_t)(r * DI + i) * DO + o] = v;
}

// ---------------- fused RGCN GEMM: out = [agg | self] @ W + bias (opt ReLU) ----------------
// A = agg [M, R*128] blocks then self [M,128]; W [(R+1)*128, NT*16]; out [M, NT*16]
// Wave computes a 16-row x (NT*16)-col stripe with V_WMMA_F32_16X16X4_F32.

template <int NT, bool RELU>
__global__ void rgcn_gemm_kernel(const float* __restrict__ agg,
                                 const float* __restrict__ self,
                                 const float* __restrict__ W,
                                 const float* __restrict__ bias,
                                 float* __restrict__ out, int M) {
  const int NDIM = NT * 16;
  int wave = threadIdx.x >> 5;
  int lane = threadIdx.x & 31;
  int row0 = (blockIdx.x * 8 + wave) * 16;
  if (row0 >= M) return;                 // wave-uniform: EXEC stays all-1s inside

  int col   = lane & 15;                 // N index within 16-tile (B/C/D layout)
  int khalf = (lane >> 4) * 2;           // K pair {0,1} or {2,3} (A/B layout)
  int arow  = row0 + col;                // A-fragment row: M = lane & 15

  v8f acc[NT];
  v8f zero = {};
#pragma unroll
  for (int t = 0; t < NT; ++t) acc[t] = zero;

  for (int r = 0; r <= R_REL; ++r) {
    const float* Ab = (r < R_REL)
        ? (agg + (size_t)arow * (R_REL * 128) + r * 128)
        : (self + (size_t)arow * 128);
    const float* Wb = W + ((size_t)r * 128 + khalf) * NDIM;
    for (int k = 0; k < 128; k += 4) {
      float2 av = *(const float2*)(Ab + k + khalf);
      v2f a; a.x = av.x; a.y = av.y;
      const float* Wk = Wb + (size_t)k * NDIM;
#pragma unroll
      for (int t = 0; t < NT; ++t) {
        v2f b;
        b.x = Wk[t * 16 + col];          // row K0   = r*128 + k + khalf
        b.y = Wk[t * 16 + col + NDIM];   // row K0+1
        acc[t] = __builtin_amdgcn_wmma_f32_16x16x4_f32(
            false, a, false, b, (short)0, acc[t], false, false);
      }
    }
  }

  int vrow0 = row0 + (lane >> 4) * 8;    // C/D layout: lanes 16-31 hold M+8
#pragma unroll
  for (int t = 0; t < NT; ++t) {
    float bb = bias[t * 16 + col];
#pragma unroll
    for (int v = 0; v < 8; ++v) {
      float val = acc[t][v] + bb;
      if (RELU) val = fmaxf(val, 0.0f);
      out[(size_t)(vrow0 + v) * NDIM + t * 16 + col] = val;
    }
  }
}

// ---------------- classifier GEMM: out = A[M,64] @ wc[64,400] + bc ----------------
// grid.y selects an 80-column group (5 tiles x 16), 5 groups cover 400.

__global__ void rgcn_cls_kernel(const float* __restrict__ A,
                                const float* __restrict__ W,     // [64,400]
                                const float* __restrict__ bias,  // [400]
                                float* __restrict__ out, int M) {
  int wave = threadIdx.x >> 5;
  int lane = threadIdx.x & 31;
  int row0 = (blockIdx.x * 8 + wave) * 16;
  if (row0 >= M) return;

  int col     = lane & 15;
  int khalf   = (lane >> 4) * 2;
  int arow    = row0 + col;
  int colbase = blockIdx.y * 80;

  v8f acc[5];
  v8f zero = {};
#pragma unroll
  for (int t = 0; t < 5; ++t) acc[t] = zero;

  for (int k = 0; k < D_OUT; k += 4) {
    float2 av = *(const float2*)(A + (size_t)arow * D_OUT + k + khalf);
    v2f a; a.x = av.x; a.y = av.y;
    const float* Wk = W + (size_t)(k + khalf) * D_ADR + colbase;
#pragma unroll
    for (int t = 0; t < 5; ++t) {
      v2f b;
      b.x = Wk[t * 16 + col];
      b.y = Wk[t * 16 + col + D_ADR];
      acc[t] = __builtin_amdgcn_wmma_f32_16x16x4_f32(
          false, a, false, b, (short)0, acc[t], false, false);
    }
  }

  int vrow0 = row0 + (lane >> 4) * 8;
#pragma unroll
  for (int t = 0; t < 5; ++t) {
    float bb = bias[colbase + t * 16 + col];
#pragma unroll
    for (int v = 0; v < 8; ++v) {
      out[(size_t)(vrow0 + v) * D_ADR + colbase + t * 16 + col] = acc[t][v] + bb;
    }
  }
}

// ---------------- host launch ----------------

extern "C" void kernel_launch(void* const* d_in, const int* in_sizes, int n_in,
                              void* d_out, int out_size, void* d_ws, size_t ws_size,
                              hipStream_t stream) {
  const float* x      = (const float*)d_in[0];
  const int*   ei     = (const int*)d_in[1];     // [2, E]
  const int*   et     = (const int*)d_in[2];     // [E]
  const float* comp1  = (const float*)d_in[3];
  const float* basis1 = (const float*)d_in[4];
  const float* root1  = (const float*)d_in[5];
  const float* bias1  = (const float*)d_in[6];
  const float* comp2  = (const float*)d_in[7];
  const float* basis2 = (const float*)d_in[8];
  const float* root2  = (const float*)d_in[9];
  const float* bias2  = (const float*)d_in[10];
  const float* wc     = (const float*)d_in[11];
  const float* bc     = (const float*)d_in[12];
  float* out = (float*)d_out;

  const int* src = ei;
  const int* dst = ei + E_EDGES;

  // workspace layout
  char* ws = (char*)d_ws;
  size_t off = 0;
  auto carve = [&](size_t bytes) {
    char* p = ws + off;
    off += (bytes + 255) & ~((size_t)255);
    return p;
  };
  const size_t aggElems = (size_t)N_NODES * R_REL * 128;
  float* agg = (float*)carve(aggElems * 4);                      // 245.8 MB (reused layer 2)
  float* cnt = (float*)carve((size_t)N_NODES * R_REL * 4);       // 1.9 MB
  float* W1s = (float*)carve((size_t)(R_REL + 1) * 128 * 128 * 4);
  float* W2s = (float*)carve((size_t)(R_REL + 1) * 128 * 64 * 4);
  float* h   = (float*)carve((size_t)N_NODES * D_HID * 4);
  float* g   = (float*)carve((size_t)N_NODES * D_OUT * 4);
  (void)ws_size; (void)in_sizes; (void)n_in; (void)out_size;

  const int E = E_EDGES, N = N_NODES;
  const int mblocks = (N + 127) / 128;   // 8 waves/block x 16 rows/wave

  // ---- layer 1 ----
  hipMemsetAsync(cnt, 0, (size_t)N * R_REL * 4, stream);
  hipMemsetAsync(agg, 0, aggElems * 4, stream);
  rgcn_count_kernel<<<(E + 255) / 256, 256, 0, stream>>>(dst, et, cnt, E);
  rgcn_scatter_kernel<<<(E + 1) / 2, 256, 0, stream>>>(x, src, dst, et, agg, E);
  rgcn_wstack_kernel<<<((R_REL + 1) * 128 * 128 + 255) / 256, 256, 0, stream>>>(
      comp1, basis1, root1, W1s, 128, 128);
  rgcn_mean_kernel<<<(int)((aggElems + 255) / 256), 256, 0, stream>>>(agg, cnt, aggElems);
  rgcn_gemm_kernel<8, true><<<mblocks, 256, 0, stream>>>(agg, x, W1s, bias1, h, N);

  // ---- layer 2 (reuse agg buffer, counts unchanged) ----
  hipMemsetAsync(agg, 0, aggElems * 4, stream);
  rgcn_scatter_kernel<<<(E + 1) / 2, 256, 0, stream>>>(h, src, dst, et, agg, E);
  rgcn_wstack_kernel<<<((R_REL + 1) * 128 * 64 + 255) / 256, 256, 0, stream>>>(
      comp2, basis2, root2, W2s, 128, 64);
  rgcn_mean_kernel<<<(int)((aggElems + 255) / 256), 256, 0, stream>>>(agg, cnt, aggElems);
  rgcn_gemm_kernel<4, false><<<mblocks, 256, 0, stream>>>(agg, h, W2s, bias2, g, N);

  // ---- classifier ----
  rgcn_cls_kernel<<<dim3(mblocks, 5), 256, 0, stream>>>(g, wc, bc, out, N);
}